// DeformableInceptionModule_55456617725982
// MI455X (gfx1250) — compile-verified
//
#include <hip/hip_runtime.h>
#include <math.h>

typedef __attribute__((ext_vector_type(2))) float v2f;
typedef __attribute__((ext_vector_type(8))) float v8f;

#define H_   80
#define W_   80
#define HW   6400
#define CIN  64
#define NB   2
#define NPOS 12800   // NB*H*W
#define NTILES 800   // NPOS/16
#define COUT_TOT 192

__device__ __forceinline__ v8f wmma_f32(v2f a, v2f b, v8f c) {
    // V_WMMA_F32_16X16X4_F32 : D = A(16x4,f32) x B(4x16,f32) + C(16x16,f32)
    return __builtin_amdgcn_wmma_f32_16x16x4_f32(
        /*neg_a=*/false, a, /*neg_b=*/false, b,
        /*c_mod=*/(short)0, c, /*reuse_a=*/false, /*reuse_b=*/false);
}

// ---------------------------------------------------------------------------
// Stage 0a: x NCHW -> NHWC  (xt[(b*HW + p)*64 + c])
// ---------------------------------------------------------------------------
__global__ void x_nhwc_kernel(const float* __restrict__ x, float* __restrict__ xt) {
    const int idx = blockIdx.x * blockDim.x + threadIdx.x;   // NB*HW*CIN total
    if (idx >= NB * HW * CIN) return;
    const int c = idx & 63;
    const int p = (idx >> 6) % HW;
    const int b = idx / (CIN * HW);
    xt[idx] = x[(b * CIN + c) * HW + p];
}

// ---------------------------------------------------------------------------
// Stage 0b: pack merged offset+mask conv weights, tap-major, zero-padded:
//   wm[m][tap*64 + c]; rows [0,2KK)=w_off, [2KK,3KK)=w_mask, rest 0.
// ---------------------------------------------------------------------------
template<int KS>
__global__ void offmask_pack_kernel(const float* __restrict__ w_off,
                                    const float* __restrict__ w_mask,
                                    float* __restrict__ wm, int mt16) {
    constexpr int KK = KS * KS;
    constexpr int K  = KK * 64;
    const int idx = blockIdx.x * blockDim.x + threadIdx.x;
    if (idx >= mt16 * K) return;
    const int m  = idx / K;
    const int r0 = idx - m * K;
    const int r  = r0 >> 6;      // tap
    const int c  = r0 & 63;      // input channel
    float v = 0.f;
    if      (m < 2 * KK) v = w_off [(m * 64 + c) * KK + r];
    else if (m < 3 * KK) v = w_mask[((m - 2 * KK) * 64 + c) * KK + r];
    wm[idx] = v;
}

// ---------------------------------------------------------------------------
// Stage 0c: w_dcn (O,C,k,k) -> wt[o][tap*64 + c]
// ---------------------------------------------------------------------------
template<int KS>
__global__ void wdcn_transpose_kernel(const float* __restrict__ w_dcn,
                                      float* __restrict__ wt) {
    constexpr int KK = KS * KS;
    const int idx = blockIdx.x * blockDim.x + threadIdx.x;
    if (idx >= 64 * 64 * KK) return;
    const int m = idx / (64 * KK);
    const int r = idx - m * (64 * KK);
    const int t = r >> 6;
    const int c = r & 63;
    wt[idx] = w_dcn[(m * 64 + c) * KK + t];
}

// ---------------------------------------------------------------------------
// Stage 1: offset+mask conv, implicit GEMM on fp32 WMMA, tap-major K.
// Branch-free inner loop: 1 float2 A load + 1 float2 B load + 1 vmul + 1 WMMA.
// ---------------------------------------------------------------------------
template<int KS>
__global__ __launch_bounds__(128)
void offmask_conv_kernel(const float* __restrict__ xt,
                         const float* __restrict__ wm,
                         const float* __restrict__ b_off, const float* __restrict__ b_mask,
                         float* __restrict__ off_out, float* __restrict__ mask_out,
                         int mtiles) {
    constexpr int KK  = KS * KS;
    constexpr int PAD = KS / 2;
    constexpr int K   = KK * 64;

    const int wave = threadIdx.x >> 5;
    const int lane = threadIdx.x & 31;
    const int half = lane >> 4;
    const int nl   = lane & 15;

    const int mtile = blockIdx.y * 4 + wave;
    if (mtile >= mtiles) return;            // wave-uniform exit

    const int m   = mtile * 16 + nl;
    const int n   = blockIdx.x * 16 + nl;
    const int b   = n / HW;
    const int rem = n - b * HW;
    const int h   = rem / W_;
    const int w   = rem - h * W_;

    const float* arow = wm + m * K;                  // zero-padded: always valid
    const float* xb   = xt + (size_t)b * HW * 64;
    const int    co   = 2 * half;                    // lane's channel sub-offset

    v8f acc = {};
    for (int ky = 0; ky < KS; ++ky) {
        const int  ih  = h - PAD + ky;
        const int  ihc = min(max(ih, 0), H_ - 1);
        const bool vy  = (ih >= 0) && (ih < H_);
        for (int kx = 0; kx < KS; ++kx) {
            const int   iw  = w - PAD + kx;
            const int   iwc = min(max(iw, 0), W_ - 1);
            const float vf  = (vy && iw >= 0 && iw < W_) ? 1.f : 0.f;
            const float* xrow = xb + (ihc * W_ + iwc) * 64;
            const float* at   = arow + (ky * KS + kx) * 64;
#pragma unroll 4
            for (int cb = 0; cb < 64; cb += 4) {
                v2f a  = *reinterpret_cast<const v2f*>(at   + cb + co);
                v2f bv = *reinterpret_cast<const v2f*>(xrow + cb + co);
                bv *= vf;                            // zero out-of-image taps
                acc = wmma_f32(a, bv, acc);
            }
        }
    }

    // Epilogue: D VGPR j, lane half -> row mtile*16 + j + 8*half, col nl.
#pragma unroll
    for (int j = 0; j < 8; ++j) {
        const int mo = mtile * 16 + j + 8 * half;
        float v = acc[j];
        if (mo < 2 * KK) {
            v += b_off[mo];
            off_out[(b * 2 * KK + mo) * HW + rem] = v;
        } else if (mo < 3 * KK) {
            const int mm = mo - 2 * KK;
            v += b_mask[mm];
            v = 1.f / (1.f + __expf(-v));
            mask_out[(b * KK + mm) * HW + rem] = v;
        }
    }
}

// ---------------------------------------------------------------------------
// Stage 2: fused bilinear gather + DCN GEMM, tap-major K, LDS-shared B tile.
// All 4 waves share the same 16 positions -> build the masked bilinear 64x16
// B tile ONCE per block per tap (cooperative, 4x fewer global gathers), then
// each wave consumes it via ds_load. LDS [c][pos] layout: conflict-free reads.
// Double-buffered with a single __syncthreads per tap.
// ---------------------------------------------------------------------------
template<int KS>
__global__ __launch_bounds__(128)
void deform_gemm_kernel(const float* __restrict__ xt,
                        const float* __restrict__ offs, const float* __restrict__ mask,
                        const float* __restrict__ wt,    // [64][KK*64]
                        float* __restrict__ out, int cbase) {
    constexpr int KK  = KS * KS;
    constexpr int PAD = KS / 2;
    constexpr int K   = KK * 64;

    __shared__ float bt[2][64][16];      // [buf][channel][position], 8 KB

    const int tid  = threadIdx.x;
    const int wave = tid >> 5;           // M tile 0..3 (64 out channels)
    const int lane = tid & 31;
    const int half = lane >> 4;
    const int nl   = lane & 15;

    // Build role: position pnl (0..15), channel group cgrp (0..7, 8 ch each).
    const int pnl  = tid & 15;
    const int cgrp = tid >> 4;

    const int n_b   = blockIdx.x * 16 + pnl;
    const int b_b   = n_b / HW;
    const int rem_b = n_b - b_b * HW;
    const int h_b   = rem_b / W_;
    const int w_b   = rem_b - h_b * W_;
    const float* xb_b = xt + (size_t)b_b * HW * 64;

    // Consume role: epilogue coordinates for this lane's column.
    const int n_c   = blockIdx.x * 16 + nl;
    const int b_c   = n_c / HW;
    const int rem_c = n_c - b_c * HW;

    const int m = wave * 16 + nl;
    const float* arow = wt + m * K;
    const int co = 2 * half;

    v8f acc = {};
    for (int ky = 0; ky < KS; ++ky) {
        for (int kx = 0; kx < KS; ++kx) {
            const int t   = ky * KS + kx;
            const int buf = t & 1;

            // ---- cooperative build of B[c][pos] for this tap ----
            {
                const float oy = offs[(b_b * 2 * KK + 2 * t    ) * HW + rem_b];
                const float ox = offs[(b_b * 2 * KK + 2 * t + 1) * HW + rem_b];
                const float mk = mask[(b_b * KK + t) * HW + rem_b];
                const float py = (float)(h_b - PAD + ky) + oy;
                const float px = (float)(w_b - PAD + kx) + ox;
                const float y0f = floorf(py), x0f = floorf(px);
                const float wy1 = py - y0f,  wx1 = px - x0f;
                const float wy0 = 1.f - wy1, wx0 = 1.f - wx1;
                const int   iy0 = (int)y0f,  ix0 = (int)x0f;

                const float* cptr[4];
                float cw[4];
#pragma unroll
                for (int q = 0; q < 4; ++q) {
                    const int yy = iy0 + (q >> 1);
                    const int xx = ix0 + (q & 1);
                    const bool valid = (yy >= 0) && (yy < H_) && (xx >= 0) && (xx < W_);
                    const int yc = min(max(yy, 0), H_ - 1);
                    const int xc = min(max(xx, 0), W_ - 1);
                    cptr[q] = xb_b + (yc * W_ + xc) * 64 + cgrp * 8;   // in-bounds
                    const float ww = ((q >> 1) ? wy1 : wy0) * ((q & 1) ? wx1 : wx0);
                    cw[q] = valid ? ww * mk : 0.f;                     // fold mask
                }
#pragma unroll
                for (int p2 = 0; p2 < 4; ++p2) {        // 4 channel pairs
                    v2f s = cw[0] * *reinterpret_cast<const v2f*>(cptr[0] + 2 * p2)
                          + cw[1] * *reinterpret_cast<const v2f*>(cptr[1] + 2 * p2)
                          + cw[2] * *reinterpret_cast<const v2f*>(cptr[2] + 2 * p2)
                          + cw[3] * *reinterpret_cast<const v2f*>(cptr[3] + 2 * p2);
                    const int c = cgrp * 8 + 2 * p2;
                    bt[buf][c    ][pnl] = s.x;
                    bt[buf][c + 1][pnl] = s.y;
                }
            }
            __syncthreads();   // one barrier per tap; double buffer covers WAR

            // ---- consume: 16 WMMAs over channel chunks of 4 ----
            const float* at = arow + t * 64;
#pragma unroll 4
            for (int cb = 0; cb < 64; cb += 4) {
                const int c4 = cb + co;
                v2f a = *reinterpret_cast<const v2f*>(at + c4);
                v2f bb;
                bb.x = bt[buf][c4    ][nl];
                bb.y = bt[buf][c4 + 1][nl];
                acc = wmma_f32(a, bb, acc);
            }
        }
    }

#pragma unroll
    for (int j = 0; j < 8; ++j) {
        const int mo = wave * 16 + j + 8 * half;           // 0..63
        out[(b_c * COUT_TOT + cbase + mo) * HW + rem_c] = acc[j];
    }
}

// ---------------------------------------------------------------------------
extern "C" void kernel_launch(void* const* d_in, const int* in_sizes, int n_in,
                              void* d_out, int out_size, void* d_ws, size_t ws_size,
                              hipStream_t stream) {
    const float* x       = (const float*)d_in[0];
    const float* w_off3  = (const float*)d_in[1];
    const float* b_off3  = (const float*)d_in[2];
    const float* w_mask3 = (const float*)d_in[3];
    const float* b_mask3 = (const float*)d_in[4];
    const float* w_dcn3  = (const float*)d_in[5];
    const float* w_off5  = (const float*)d_in[6];
    const float* b_off5  = (const float*)d_in[7];
    const float* w_mask5 = (const float*)d_in[8];
    const float* b_mask5 = (const float*)d_in[9];
    const float* w_dcn5  = (const float*)d_in[10];
    const float* w_off7  = (const float*)d_in[11];
    const float* b_off7  = (const float*)d_in[12];
    const float* w_mask7 = (const float*)d_in[13];
    const float* b_mask7 = (const float*)d_in[14];
    const float* w_dcn7  = (const float*)d_in[15];
    float* out = (float*)d_out;

    // Workspace carve-up (floats), ~20 MB total:
    float* ws   = (float*)d_ws;
    float* xt   = ws;                        // 2*6400*64 = 819200
    float* wm3  = xt  + NB * HW * CIN;       // 32*576    = 18432
    float* wm5  = wm3 + 32  * 576;           // 80*1600   = 128000
    float* wm7  = wm5 + 80  * 1600;          // 160*3136  = 501760
    float* wt3  = wm7 + 160 * 3136;          // 64*576    = 36864
    float* wt5  = wt3 + 64 * 576;            // 64*1600   = 102400
    float* wt7  = wt5 + 64 * 1600;           // 64*3136   = 200704
    float* off3 = wt7 + 64 * 3136;           // 2*18*6400 = 230400
    float* msk3 = off3 + 2 * 18 * HW;        // 2*9*6400  = 115200
    float* off5 = msk3 + 2 * 9  * HW;        // 2*50*6400 = 640000
    float* msk5 = off5 + 2 * 50 * HW;        // 2*25*6400 = 320000
    float* off7 = msk5 + 2 * 25 * HW;        // 2*98*6400 = 1254400
    float* msk7 = off7 + 2 * 98 * HW;        // 2*49*6400 = 627200

    // Stage 0: layout transforms (tiny, one-time)
    x_nhwc_kernel<<<(NB * HW * CIN + 255) / 256, 256, 0, stream>>>(x, xt);
    offmask_pack_kernel<3><<<(32  * 576  + 255) / 256, 256, 0, stream>>>(w_off3, w_mask3, wm3, 32);
    offmask_pack_kernel<5><<<(80  * 1600 + 255) / 256, 256, 0, stream>>>(w_off5, w_mask5, wm5, 80);
    offmask_pack_kernel<7><<<(160 * 3136 + 255) / 256, 256, 0, stream>>>(w_off7, w_mask7, wm7, 160);
    wdcn_transpose_kernel<3><<<(64 * 64 * 9  + 255) / 256, 256, 0, stream>>>(w_dcn3, wt3);
    wdcn_transpose_kernel<5><<<(64 * 64 * 25 + 255) / 256, 256, 0, stream>>>(w_dcn5, wt5);
    wdcn_transpose_kernel<7><<<(64 * 64 * 49 + 255) / 256, 256, 0, stream>>>(w_dcn7, wt7);

    dim3 blk(128);
    // Stage 1: offset/mask convs. Mtiles = 2, 5, 10.
    offmask_conv_kernel<3><<<dim3(NTILES, 1), blk, 0, stream>>>(xt, wm3, b_off3, b_mask3, off3, msk3, 2);
    offmask_conv_kernel<5><<<dim3(NTILES, 2), blk, 0, stream>>>(xt, wm5, b_off5, b_mask5, off5, msk5, 5);
    offmask_conv_kernel<7><<<dim3(NTILES, 3), blk, 0, stream>>>(xt, wm7, b_off7, b_mask7, off7, msk7, 10);

    // Stage 2: fused deformable gather + GEMM, concat via channel base.
    deform_gemm_kernel<3><<<NTILES, blk, 0, stream>>>(xt, off3, msk3, wt3, out, 0);
    deform_gemm_kernel<5><<<NTILES, blk, 0, stream>>>(xt, off5, msk5, wt5, out, 64);
    deform_gemm_kernel<7><<<NTILES, blk, 0, stream>>>(xt, off7, msk7, wt7, out, 128);
}